// GNN_Layer_68367289418123
// MI455X (gfx1250) — compile-verified
//
#include <hip/hip_runtime.h>

typedef float v2f __attribute__((ext_vector_type(2)));
typedef float v8f __attribute__((ext_vector_type(8)));

#define DIM 128

// ---------------------------------------------------------------------------
// Kernel 1: dual GEMM via V_WMMA_F32_16X16X4_F32.
//   out[m][n] = sum_k X[m][k]*Wself[n][k] + bias[n]      (h_self + bias)
//   xw [m][n] = sum_k X[m][k]*Wneigh[n][k]               (per-node neighbor msg)
// One wave computes one 16x16 tile of BOTH outputs (shared A-matrix loads).
// Block = 256 threads = 8 waves; wave w handles N-tile w (8 tiles * 16 = 128).
// Grid.x = n_nodes/16 M-tiles (50000 is an exact multiple of 16).
//
// WMMA f32 16x16x4 operand layout (wave32):
//   A (16x4, MxK):  lane L<16  -> row m0+L,  VGPR0/1 = K=k+0,k+1
//                   lane L>=16 -> row m0+L-16, VGPR0/1 = K=k+2,k+3
//   B (4x16, KxN):  lane L<16  -> col n0+L,  VGPR0/1 = K=k+0,k+1
//                   lane L>=16 -> col n0+L-16, VGPR0/1 = K=k+2,k+3
//   (B[k][n] = W[n][k], so both A and B are a contiguous float2 at
//    row = base + (lane&15), col = k + 2*(lane>>4).)
//   C/D: VGPR j: lanes 0-15 -> (m0+j, n0+lane); lanes 16-31 -> (m0+j+8, ...).
// ---------------------------------------------------------------------------
__global__ __launch_bounds__(256) void gnn_gemm_wmma(
    const float* __restrict__ X,
    const float* __restrict__ Wself,
    const float* __restrict__ Wneigh,
    const float* __restrict__ bias,
    float* __restrict__ out,   // [n_nodes][DIM] : h_self + bias
    float* __restrict__ xw)    // [n_nodes][DIM] : X @ Wneigh^T
{
    const int lane    = threadIdx.x & 31;
    const int wave    = threadIdx.x >> 5;      // 0..7 -> N-tile
    const int m0      = blockIdx.x * 16;
    const int n0      = wave * 16;
    const int r       = lane & 15;
    const int halfSel = lane >> 4;             // 0 or 1
    const int kOff    = 2 * halfSel;

    const float* xRow  = X      + (size_t)(m0 + r) * DIM + kOff;
    const float* wsRow = Wself  + (size_t)(n0 + r) * DIM + kOff;
    const float* wnRow = Wneigh + (size_t)(n0 + r) * DIM + kOff;

    v8f c_self  = {};
    v8f c_neigh = {};

#pragma unroll 8
    for (int k = 0; k < DIM; k += 4) {
        v2f a  = *(const v2f*)(xRow  + k);
        v2f bs = *(const v2f*)(wsRow + k);
        v2f bn = *(const v2f*)(wnRow + k);
        c_self  = __builtin_amdgcn_wmma_f32_16x16x4_f32(
                      false, a, false, bs, (short)0, c_self,  false, false);
        c_neigh = __builtin_amdgcn_wmma_f32_16x16x4_f32(
                      false, a, false, bn, (short)0, c_neigh, false, false);
    }

    const int   col = n0 + r;
    const float bv  = bias[col];

#pragma unroll
    for (int j = 0; j < 8; ++j) {
        const int row = m0 + j + 8 * halfSel;
        out[(size_t)row * DIM + col] = c_self[j] + bv;
        xw [(size_t)row * DIM + col] = c_neigh[j];
    }
}

// ---------------------------------------------------------------------------
// Kernel 2: edge scatter.  One wave per edge; lane l covers 4 contiguous
// floats.  xw (25.6 MB) and out (25.6 MB) both fit in the 192 MB L2, so the
// 410 MB of gather traffic and the fp32 atomics stay on-chip.
// ---------------------------------------------------------------------------
__global__ __launch_bounds__(256) void gnn_edge_scatter(
    const int* __restrict__ src,
    const int* __restrict__ dst,
    const float* __restrict__ xw,
    float* __restrict__ out,
    int n_edges)
{
    const int e = blockIdx.x * 8 + (threadIdx.x >> 5);
    if (e >= n_edges) return;
    const int lane = threadIdx.x & 31;

    const int s = src[e];
    const int d = dst[e];

    const float4 v = *(const float4*)(xw + (size_t)s * DIM + lane * 4);
    float* o = out + (size_t)d * DIM + lane * 4;
    atomicAdd(o + 0, v.x);
    atomicAdd(o + 1, v.y);
    atomicAdd(o + 2, v.z);
    atomicAdd(o + 3, v.w);
}

// ---------------------------------------------------------------------------
// Kernel 3: in-place ReLU, float4 vectorized.
// ---------------------------------------------------------------------------
__global__ __launch_bounds__(256) void gnn_relu(float* __restrict__ out, int n4)
{
    const int i = blockIdx.x * blockDim.x + threadIdx.x;
    if (i >= n4) return;
    float4 v = ((float4*)out)[i];
    v.x = fmaxf(v.x, 0.0f);
    v.y = fmaxf(v.y, 0.0f);
    v.z = fmaxf(v.z, 0.0f);
    v.w = fmaxf(v.w, 0.0f);
    ((float4*)out)[i] = v;
}

// ---------------------------------------------------------------------------
// Inputs (setup_inputs order): X, src, dst, W_self, W_neigh, bias
// d_out: [n_nodes][128] float32.  d_ws: xw scratch (needs n_nodes*128*4 B).
// ---------------------------------------------------------------------------
extern "C" void kernel_launch(void* const* d_in, const int* in_sizes, int n_in,
                              void* d_out, int out_size, void* d_ws, size_t ws_size,
                              hipStream_t stream)
{
    const float* X      = (const float*)d_in[0];
    const int*   src    = (const int*)  d_in[1];   // harness delivers indices as int32
    const int*   dst    = (const int*)  d_in[2];
    const float* Wself  = (const float*)d_in[3];
    const float* Wneigh = (const float*)d_in[4];
    const float* bias   = (const float*)d_in[5];

    float* out = (float*)d_out;
    float* xw  = (float*)d_ws;

    const int n_nodes = in_sizes[0] / DIM;   // 50000
    const int n_edges = in_sizes[1];         // 800000

    // Phase 1: h_self + bias -> out,  X @ Wneigh^T -> xw
    const int mtiles = n_nodes / 16;         // 3125 (exact)
    gnn_gemm_wmma<<<mtiles, 256, 0, stream>>>(X, Wself, Wneigh, bias, out, xw);

    // Phase 2: scatter-add neighbor messages into out
    const int eblocks = (n_edges + 7) / 8;   // 8 waves (edges) per block
    gnn_edge_scatter<<<eblocks, 256, 0, stream>>>(src, dst, xw, out, n_edges);

    // Phase 3: ReLU in place
    const int n4 = (n_nodes * DIM) / 4;
    gnn_relu<<<(n4 + 255) / 256, 256, 0, stream>>>(out, n4);
}